// SOCA_12584254177576
// MI455X (gfx1250) — compile-verified
//
#include <hip/hip_runtime.h>
#include <math.h>

typedef __attribute__((ext_vector_type(2))) float v2f;
typedef __attribute__((ext_vector_type(8))) float v8f;
typedef int v4i __attribute__((vector_size(16)));

#define BATCH 16
#define CH    64
#define MPIX  4096                 // 64*64 spatial
#define KSPLIT 8
#define KCHUNK (MPIX / KSPLIT)     // 512
#define KSTAGE 64
#define TS     68                  // LDS row stride (floats): 16B aligned, bank-skewed

// workspace layout in floats
#define WS_COV 0
#define WS_SUM (BATCH * CH * CH)          // 65536
#define WS_SIG (WS_SUM + BATCH * CH)      // 66560

// ---- gfx1250 async global->LDS path (guarded; fallback = float4 staging) ----
#if defined(__has_builtin)
#  if __has_builtin(__builtin_amdgcn_global_load_async_to_lds_b128)
#    define SOCA_ASYNC 1
#  endif
#  if __has_builtin(__builtin_amdgcn_s_wait_asynccnt)
#    define SOCA_HAVE_WAIT_ASYNC 1
#  endif
#endif
#ifndef SOCA_ASYNC
#  define SOCA_ASYNC 0
#endif

typedef __attribute__((address_space(1))) v4i* gptr4;   // global int32x4*
typedef __attribute__((address_space(3))) v4i* lptr4;   // LDS int32x4*

__device__ __forceinline__ void soca_wait_async() {
#if defined(SOCA_HAVE_WAIT_ASYNC)
    __builtin_amdgcn_s_wait_asynccnt(0);
#else
    asm volatile("s_wait_asynccnt 0x0" ::: "memory");
#endif
}

// ---------------------------------------------------------------- zero scratch
__global__ void soca_zero(float* ws, int n) {
    int i = blockIdx.x * blockDim.x + threadIdx.x;
    if (i < n) ws[i] = 0.0f;
}

// ------------------------------------------------------- Gram matrix (WMMA f32)
// G[b] = xf[b] @ xf[b]^T accumulated over K-splits; s[b][c] = channel sums.
// 8 wave32 waves: wave pair (2w, 2w+1) owns one 2x2 block of 16x16 tiles and
// splits the staged K range; partials combined through LDS at chunk end.
__global__ __launch_bounds__(256) void soca_gram(const float* __restrict__ x,
                                                 float* __restrict__ ws) {
    __shared__ float tile[CH * TS];     // 4352 floats; doubles as combine scratch

    const int b    = blockIdx.x;
    const int kc   = blockIdx.y;
    const int tid  = threadIdx.x;
    const int wave = tid >> 5;
    const int lane = tid & 31;
    const int lm   = lane & 15;
    const int lh   = lane >> 4;

    const int bw = wave >> 1;           // 2x2 tile block 0..3
    const int kh = wave & 1;            // K-half within stage
    const int br = bw >> 1, bcn = bw & 1;
    const int ti0 = br * 2, ti1 = br * 2 + 1;
    const int tj0 = bcn * 2, tj1 = bcn * 2 + 1;

    const float* xb = x + (size_t)b * CH * MPIX;

    v8f acc00 = {}, acc01 = {}, acc10 = {}, acc11 = {};
    float csum = 0.0f;

    for (int s = 0; s < KCHUNK / KSTAGE; ++s) {
        const int k0 = kc * KCHUNK + s * KSTAGE;

        // stage 64 rows x 64 cols fp32 into LDS
#if SOCA_ASYNC
        for (int f = tid; f < CH * (KSTAGE / 4); f += 256) {
            const int row = f >> 4;
            const int c4  = f & 15;
            __builtin_amdgcn_global_load_async_to_lds_b128(
                (gptr4)(xb + (size_t)row * MPIX + k0 + c4 * 4),
                (lptr4)&tile[row * TS + c4 * 4], 0, 0);
        }
        soca_wait_async();
#else
        for (int f = tid; f < CH * (KSTAGE / 4); f += 256) {
            const int row = f >> 4;
            const int c4  = f & 15;
            float4 v = *reinterpret_cast<const float4*>(xb + (size_t)row * MPIX + k0 + c4 * 4);
            *reinterpret_cast<float4*>(&tile[row * TS + c4 * 4]) = v;
            __builtin_prefetch(xb + (size_t)row * MPIX + k0 + KSTAGE + c4 * 4, 0, 0);
        }
#endif
        __syncthreads();

        // this wave's half of the K stage: 8 steps x 4 WMMA, 4 frag loads/step
        for (int kk = kh * 32; kk < kh * 32 + 32; kk += 4) {
            const int col = kk + lh * 2;
            v2f a0, a1, b0, b1;
            a0.x = tile[(ti0 * 16 + lm) * TS + col];
            a0.y = tile[(ti0 * 16 + lm) * TS + col + 1];
            a1.x = tile[(ti1 * 16 + lm) * TS + col];
            a1.y = tile[(ti1 * 16 + lm) * TS + col + 1];
            b0.x = tile[(tj0 * 16 + lm) * TS + col];
            b0.y = tile[(tj0 * 16 + lm) * TS + col + 1];
            b1.x = tile[(tj1 * 16 + lm) * TS + col];
            b1.y = tile[(tj1 * 16 + lm) * TS + col + 1];
            acc00 = __builtin_amdgcn_wmma_f32_16x16x4_f32(false, a0, false, b0,
                                                          (short)0, acc00, false, false);
            acc01 = __builtin_amdgcn_wmma_f32_16x16x4_f32(false, a0, false, b1,
                                                          (short)0, acc01, false, false);
            acc10 = __builtin_amdgcn_wmma_f32_16x16x4_f32(false, a1, false, b0,
                                                          (short)0, acc10, false, false);
            acc11 = __builtin_amdgcn_wmma_f32_16x16x4_f32(false, a1, false, b1,
                                                          (short)0, acc11, false, false);
        }

        // per-channel partial sums over the full stage (waves 0-1, wave-uniform)
        if (tid < CH) {
            float ssum = 0.0f;
            for (int j = 0; j < KSTAGE; ++j) ssum += tile[tid * TS + j];
            csum += ssum;
        }
        __syncthreads();
    }

    // combine the two K-half partials of each block through LDS scratch
    float* scratch = tile;              // 4 blocks * 4 tiles * 256 = 4096 floats
    if (kh == 1) {
        for (int r = 0; r < 8; ++r) {
            scratch[bw * 1024 +   0 + r * 32 + lane] = acc00[r];
            scratch[bw * 1024 + 256 + r * 32 + lane] = acc01[r];
            scratch[bw * 1024 + 512 + r * 32 + lane] = acc10[r];
            scratch[bw * 1024 + 768 + r * 32 + lane] = acc11[r];
        }
    }
    __syncthreads();
    if (kh == 0) {
        float* cov = ws + WS_COV + (size_t)b * CH * CH;
        for (int r = 0; r < 8; ++r) {
            const int m0 = ti0 * 16 + lh * 8 + r;
            const int m1 = ti1 * 16 + lh * 8 + r;
            const int n0 = tj0 * 16 + lm;
            const int n1 = tj1 * 16 + lm;
            atomicAdd(&cov[m0 * CH + n0], acc00[r] + scratch[bw * 1024 +   0 + r * 32 + lane]);
            atomicAdd(&cov[m0 * CH + n1], acc01[r] + scratch[bw * 1024 + 256 + r * 32 + lane]);
            atomicAdd(&cov[m1 * CH + n0], acc10[r] + scratch[bw * 1024 + 512 + r * 32 + lane]);
            atomicAdd(&cov[m1 * CH + n1], acc11[r] + scratch[bw * 1024 + 768 + r * 32 + lane]);
        }
    }
    if (tid < CH) atomicAdd(ws + WS_SUM + b * CH + tid, csum);
}

// --------------------------------------------------- LDS-resident WMMA helpers
__device__ __forceinline__ v8f lds_mm_tile(const float* A, const float* B,
                                           int ti, int tj, int lm, int lh) {
    v8f acc = {};
    for (int kk = 0; kk < CH; kk += 4) {
        const int col = kk + lh * 2;
        v2f a, b;
        a.x = A[(ti * 16 + lm) * TS + col];
        a.y = A[(ti * 16 + lm) * TS + col + 1];
        b.x = B[col * TS + tj * 16 + lm];
        b.y = B[(col + 1) * TS + tj * 16 + lm];
        acc = __builtin_amdgcn_wmma_f32_16x16x4_f32(false, a, false, b,
                                                    (short)0, acc, false, false);
    }
    return acc;
}

__device__ __forceinline__ void store_tile(float* M, v8f acc, int ti, int tj,
                                           int lm, int lh) {
    for (int r = 0; r < 8; ++r)
        M[(ti * 16 + lh * 8 + r) * TS + tj * 16 + lm] = acc[r];
}

// element = diagval*(row==col) + scale*acc   (for 1.5I - 0.5P and 3I - P forms)
__device__ __forceinline__ void store_affine(float* M, v8f acc, int ti, int tj,
                                             int lm, int lh, float diagval, float scale) {
    for (int r = 0; r < 8; ++r) {
        const int row = ti * 16 + lh * 8 + r;
        const int col = tj * 16 + lm;
        M[row * TS + col] = (row == col ? diagval : 0.0f) + scale * acc[r];
    }
}

// ------------------------- finalize cov + Newton-Schulz sqrt + FC head (per b)
__global__ __launch_bounds__(256) void soca_ns(const float* __restrict__ w1,
                                               const float* __restrict__ b1,
                                               const float* __restrict__ w2,
                                               const float* __restrict__ b2,
                                               float* __restrict__ ws) {
    __shared__ float matY[CH * TS];
    __shared__ float matZ[CH * TS];
    __shared__ float matT[CH * TS];
    __shared__ float red[256];
    __shared__ float desc[CH];
    __shared__ float hid[CH];

    const int b    = blockIdx.x;
    const int tid  = threadIdx.x;
    const int wave = tid >> 5;
    const int lane = tid & 31;
    const int lm   = lane & 15;
    const int lh   = lane >> 4;
    const int t0 = wave, t1 = wave + 8;
    const int ti0 = t0 >> 2, tj0 = t0 & 3;
    const int ti1 = t1 >> 2, tj1 = t1 & 3;

    const float* G    = ws + WS_COV + (size_t)b * CH * CH;
    const float* sums = ws + WS_SUM + b * CH;
    const float minv  = 1.0f / (float)MPIX;

    // cov = (G - s s^T / M) / M  -> matY ; normA = sum(cov)
    float part = 0.0f;
    for (int i = tid; i < CH * CH; i += 256) {
        const int r = i >> 6, c = i & 63;
        const float v = (G[i] - sums[r] * sums[c] * minv) * minv;
        matY[r * TS + c] = v;
        part += v;
    }
    red[tid] = part;
    __syncthreads();
    for (int s = 128; s > 0; s >>= 1) {
        if (tid < s) red[tid] += red[tid + s];
        __syncthreads();
    }
    const float normA = red[0];
    const float inorm = 1.0f / normA;
    __syncthreads();

    // A = cov/normA (in matY); T = 1.5I - 0.5A (matT); Z = T (matZ)
    for (int i = tid; i < CH * CH; i += 256) {
        const int r = i >> 6, c = i & 63;
        const float a = matY[r * TS + c] * inorm;
        matY[r * TS + c] = a;
        const float t = (r == c ? 1.5f : 0.0f) - 0.5f * a;
        matT[r * TS + c] = t;
        matZ[r * TS + c] = t;
    }
    __syncthreads();

    // Y = A @ T
    {
        v8f y0 = lds_mm_tile(matY, matT, ti0, tj0, lm, lh);
        v8f y1 = lds_mm_tile(matY, matT, ti1, tj1, lm, lh);
        __syncthreads();
        store_tile(matY, y0, ti0, tj0, lm, lh);
        store_tile(matY, y1, ti1, tj1, lm, lh);
        __syncthreads();
    }

    // middle iterations: T = 1.5I - 0.5 Z@Y ; Y = Y@T ; Z = T@Z
    for (int it = 0; it < 3; ++it) {
        v8f p0 = lds_mm_tile(matZ, matY, ti0, tj0, lm, lh);
        v8f p1 = lds_mm_tile(matZ, matY, ti1, tj1, lm, lh);
        store_affine(matT, p0, ti0, tj0, lm, lh, 1.5f, -0.5f);
        store_affine(matT, p1, ti1, tj1, lm, lh, 1.5f, -0.5f);
        __syncthreads();
        v8f yn0 = lds_mm_tile(matY, matT, ti0, tj0, lm, lh);
        v8f yn1 = lds_mm_tile(matY, matT, ti1, tj1, lm, lh);
        v8f zn0 = lds_mm_tile(matT, matZ, ti0, tj0, lm, lh);
        v8f zn1 = lds_mm_tile(matT, matZ, ti1, tj1, lm, lh);
        __syncthreads();
        store_tile(matY, yn0, ti0, tj0, lm, lh);
        store_tile(matY, yn1, ti1, tj1, lm, lh);
        store_tile(matZ, zn0, ti0, tj0, lm, lh);
        store_tile(matZ, zn1, ti1, tj1, lm, lh);
        __syncthreads();
    }

    // final: T' = 3I - Z@Y ; F = Y@T' (the 0.5 and sqrt(normA) fold into desc)
    {
        v8f p0 = lds_mm_tile(matZ, matY, ti0, tj0, lm, lh);
        v8f p1 = lds_mm_tile(matZ, matY, ti1, tj1, lm, lh);
        store_affine(matT, p0, ti0, tj0, lm, lh, 3.0f, -1.0f);
        store_affine(matT, p1, ti1, tj1, lm, lh, 3.0f, -1.0f);
        __syncthreads();
        v8f f0 = lds_mm_tile(matY, matT, ti0, tj0, lm, lh);
        v8f f1 = lds_mm_tile(matY, matT, ti1, tj1, lm, lh);
        __syncthreads();
        store_tile(matT, f0, ti0, tj0, lm, lh);
        store_tile(matT, f1, ti1, tj1, lm, lh);
        __syncthreads();
    }

    // channel descriptor: mean over rows, scaled by 0.5*sqrt(normA)
    if (tid < CH) {
        float s = 0.0f;
        for (int m = 0; m < CH; ++m) s += matT[m * TS + tid];
        desc[tid] = s * (0.5f * sqrtf(normA) / (float)CH);
    }
    __syncthreads();

    // conv1 (center tap of 3x3 on 1x1 input) + relu
    if (tid < CH) {
        float a = b1[tid];
        for (int i = 0; i < CH; ++i) a += w1[tid * CH * 9 + i * 9 + 4] * desc[i];
        hid[tid] = fmaxf(a, 0.0f);
    }
    __syncthreads();
    // conv2 + sigmoid
    if (tid < CH) {
        float a = b2[tid];
        for (int i = 0; i < CH; ++i) a += w2[tid * CH * 9 + i * 9 + 4] * hid[i];
        ws[WS_SIG + b * CH + tid] = 1.0f / (1.0f + __expf(-a));
    }
}

// ----------------------------------------------------------- broadcast scaling
__global__ __launch_bounds__(256) void soca_scale(const float* __restrict__ x,
                                                  const float* __restrict__ ws,
                                                  float* __restrict__ out) {
    const size_t f = (size_t)blockIdx.x * blockDim.x + threadIdx.x; // float4 idx
    const size_t e = f * 4;
    const int bc = (int)(e >> 12);         // / 4096
    const int c = bc & 63;
    const int b = bc >> 6;
    const float s = ws[WS_SIG + b * CH + c];
    float4 v = reinterpret_cast<const float4*>(x)[f];
    float4 o;
    o.x = v.x * s; o.y = v.y * s; o.z = v.z * s; o.w = v.w * s;
    reinterpret_cast<float4*>(out)[f] = o;
}

// ------------------------------------------------------------------- launcher
extern "C" void kernel_launch(void* const* d_in, const int* in_sizes, int n_in,
                              void* d_out, int out_size, void* d_ws, size_t ws_size,
                              hipStream_t stream) {
    (void)in_sizes; (void)n_in; (void)out_size; (void)ws_size;
    const float* x  = (const float*)d_in[0];
    const float* w1 = (const float*)d_in[1];
    const float* b1 = (const float*)d_in[2];
    const float* w2 = (const float*)d_in[3];
    const float* b2 = (const float*)d_in[4];
    float* ws  = (float*)d_ws;
    float* out = (float*)d_out;

    const int nzero = BATCH * CH * CH + BATCH * CH;   // cov + sums
    soca_zero<<<(nzero + 255) / 256, 256, 0, stream>>>(ws, nzero);
    soca_gram<<<dim3(BATCH, KSPLIT), 256, 0, stream>>>(x, ws);
    soca_ns<<<BATCH, 256, 0, stream>>>(w1, b1, w2, b2, ws);
    const int n4 = BATCH * CH * MPIX / 4;
    soca_scale<<<(n4 + 255) / 256, 256, 0, stream>>>(x, ws, out);
}